// Model_48911087567432
// MI455X (gfx1250) — compile-verified
//
#include <hip/hip_runtime.h>
#include <stdint.h>

// Embedding backward (scatter-add) for MI455X / gfx1250.
// grad_output: [N, D] f32 (N = B*S = 131072, D = 128)
// indices:     [N] int64, PADDING_IDX = 0 skipped
// out:         [V, D] f32 (V = out_size / D = 200000)
//
// Memory/atomic-bound op (0.07 FLOP/byte): no matmul content, so no WMMA.
// MI455X-specific leverage:
//  - 192MB L2 holds the entire 102MB output table -> scatter runs at L2 rate
//  - native global_atomic_add_f32 (no-return, fire-and-forget)
//  - non-temporal (TH=NT) streaming loads of grad keep L2 for the table
//  - wave32: D=128 f32 == 32 lanes x float4 -> one wave per row/token
//
// Pass structure (counts array in d_ws):
//  1) zero counts
//  2) count contributors per row (u32 atomics, ~131K, negligible)
//  3) conditional row-zero: skip rows with count==1 (fully overwritten by the
//     sole contributor's stores) -> saves ~34% of the 97MB zero traffic
//  4) scatter: count==1 -> plain b128 stores (exact); else f32 atomics

typedef float f32x4 __attribute__((ext_vector_type(4)));
typedef int   i32x4 __attribute__((ext_vector_type(4)));

#define PADDING_IDX 0LL

// ---------------------------------------------------------------- zero counts
__global__ __launch_bounds__(256)
void eb_zero_counts_kernel(int* __restrict__ counts, long long n_cnt) {
    long long tid    = (long long)blockIdx.x * blockDim.x + threadIdx.x;
    long long stride = (long long)gridDim.x * blockDim.x;
    long long n4 = n_cnt >> 2;
    i32x4* c4 = (i32x4*)counts;
    i32x4 z = {0, 0, 0, 0};
    for (long long i = tid; i < n4; i += stride) c4[i] = z;
    if (tid == 0)
        for (long long i = n4 << 2; i < n_cnt; ++i) counts[i] = 0;
}

// --------------------------------------------------------- unconditional zero
// (fallback path when no workspace for counts)
__global__ __launch_bounds__(256)
void eb_zero_out_kernel(float* __restrict__ out, long long n_out) {
    long long tid    = (long long)blockIdx.x * blockDim.x + threadIdx.x;
    long long stride = (long long)gridDim.x * blockDim.x;
    long long n4 = n_out >> 2;
    f32x4* o4 = (f32x4*)out;
    f32x4 z = {0.f, 0.f, 0.f, 0.f};
    for (long long i = tid; i < n4; i += stride) o4[i] = z;
    if (tid == 0)
        for (long long i = n4 << 2; i < n_out; ++i) out[i] = 0.f;
}

// ------------------------------------------------------------------ count
__global__ __launch_bounds__(256)
void eb_count_kernel(const long long* __restrict__ idx, int n_tok,
                     long long V, int* __restrict__ counts) {
    int t = blockIdx.x * blockDim.x + threadIdx.x;
    if (t >= n_tok) return;
    long long v = idx[t];
    if (v != PADDING_IDX && v >= 0 && v < V)
        atomicAdd(&counts[v], 1);   // global_atomic_add_u32, no return
}

// ---------------------------------------------------- conditional row zeroing
// One wave32 per embedding row; skip rows with exactly one contributor
// (those are fully overwritten by the scatter's store path).
__global__ __launch_bounds__(256)
void eb_row_zero_kernel(float* __restrict__ out,
                        const int* __restrict__ counts,
                        long long V, int D) {
    const int waves_per_blk = blockDim.x >> 5;
    const long long row = (long long)blockIdx.x * waves_per_blk
                        + (threadIdx.x >> 5);
    const int lane = threadIdx.x & 31;
    if (row >= V) return;
    if (counts[row] == 1) return;         // sole writer will store whole row

    float* __restrict__ r = out + row * (long long)D;
    if ((D & 3) == 0) {
        const int D4 = D >> 2;
        f32x4 z = {0.f, 0.f, 0.f, 0.f};
        f32x4* r4 = (f32x4*)r;
        for (int j = lane; j < D4; j += 32) r4[j] = z;
    } else {
        for (int j = lane; j < D; j += 32) r[j] = 0.f;
    }
}

// ------------------------------------------------------------------ scatter
// One wave32 per token. count==1 -> exact full-row vector stores,
// otherwise global_atomic_add_f32 per element.
__global__ __launch_bounds__(256)
void eb_scatter_kernel(const float* __restrict__ grad,
                       const long long* __restrict__ idx,
                       const int* __restrict__ counts,
                       float* __restrict__ out,
                       int n_tok, int D, long long V) {
    const int waves_per_blk = blockDim.x >> 5;
    const int wave = blockIdx.x * waves_per_blk + (threadIdx.x >> 5);
    const int lane = threadIdx.x & 31;
    if (wave >= n_tok) return;

    long long v = idx[wave];              // 32 lanes, same addr: one request
    if (v == PADDING_IDX || v < 0 || v >= V) return;

    const int c = counts ? counts[v] : 2; // no counts buffer -> atomic path

    const float* __restrict__ src = grad + (long long)wave * D;
    float* __restrict__ dst = out + v * (long long)D;

    if ((D & 3) == 0) {
        const int D4 = D >> 2;
        const f32x4* __restrict__ s4 = (const f32x4*)src;
        for (int j = lane; j < D4; j += 32) {
            f32x4 val = __builtin_nontemporal_load(&s4[j]);  // streamed once
            float* p = dst + 4 * j;
            if (c == 1) {
                *(f32x4*)p = val;         // exact, full-width store
            } else {
                atomicAdd(p + 0, val.x);  // global_atomic_add_f32 in L2
                atomicAdd(p + 1, val.y);
                atomicAdd(p + 2, val.z);
                atomicAdd(p + 3, val.w);
            }
        }
    } else {
        for (int j = lane; j < D; j += 32) {
            float val = __builtin_nontemporal_load(&src[j]);
            if (c == 1) dst[j] = val;
            else        atomicAdd(&dst[j], val);
        }
    }
}

// ------------------------------------------------------------------ launch
extern "C" void kernel_launch(void* const* d_in, const int* in_sizes, int n_in,
                              void* d_out, int out_size, void* d_ws, size_t ws_size,
                              hipStream_t stream) {
    const float*     grad = (const float*)d_in[0];
    const long long* idx  = (const long long*)d_in[1];
    // d_in[2] (num_embeddings) lives on device; derive V on host instead.
    const int       n_tok = in_sizes[1];               // B*S
    const int       D     = in_sizes[0] / in_sizes[1]; // 128
    const long long V     = (long long)out_size / D;   // 200000
    float* out = (float*)d_out;

    int* counts = (ws_size >= (size_t)V * sizeof(int)) ? (int*)d_ws : nullptr;

    const int waves_per_blk = 8;          // 256 threads = 8 wave32

    if (counts) {
        // 1) zero counts (800 KB)
        {
            long long n4 = V >> 2;
            int grid = (int)((n4 + 255) / 256);
            if (grid < 1) grid = 1;
            if (grid > 16384) grid = 16384;
            eb_zero_counts_kernel<<<grid, 256, 0, stream>>>(counts, V);
        }
        // 2) per-row contributor counts
        {
            int grid = (n_tok + 255) / 256;
            eb_count_kernel<<<grid, 256, 0, stream>>>(idx, n_tok, V, counts);
        }
        // 3) zero only rows not fully overwritten (count != 1)
        {
            long long waves = V;
            int grid = (int)((waves + waves_per_blk - 1) / waves_per_blk);
            eb_row_zero_kernel<<<grid, 256, 0, stream>>>(out, counts, V, D);
        }
    } else {
        // Fallback: zero everything, all-atomic scatter
        long long n4 = (long long)out_size >> 2;
        int grid = (int)((n4 + 255) / 256);
        if (grid < 1) grid = 1;
        if (grid > 65535) grid = 65535;
        eb_zero_out_kernel<<<grid, 256, 0, stream>>>(out, (long long)out_size);
    }

    // 4) scatter: one wave32 per token
    {
        int grid = (n_tok + waves_per_blk - 1) / waves_per_blk;
        eb_scatter_kernel<<<grid, 256, 0, stream>>>(grad, idx, counts, out,
                                                    n_tok, D, V);
    }
}